// ConvolutionalBlockAttentionModule_62371515072795
// MI455X (gfx1250) — compile-verified
//
#include <hip/hip_runtime.h>
#include <hip/hip_bf16.h>
#include <math.h>

// ---------------- problem constants ----------------
#define Bn   32
#define Hn   56
#define Wn   56
#define Cn   256
#define HWn  (Hn * Wn)          // 3136
#define HID  16                 // C / R
#define NCHK 8                  // HW reduction chunks per batch
#define CHUNK (HWn / NCHK)      // 392

typedef float v2f __attribute__((ext_vector_type(2)));
typedef float v8f __attribute__((ext_vector_type(8)));

__device__ __forceinline__ v8f wmma4(v2f a, v2f b, v8f c) {
    // V_WMMA_F32_16X16X4_F32 : D = A(16x4) * B(4x16) + C(16x16)
    return __builtin_amdgcn_wmma_f32_16x16x4_f32(
        /*neg_a=*/false, a, /*neg_b=*/false, b,
        /*c_mod=*/(short)0, c, /*reuse_a=*/false, /*reuse_b=*/false);
}

__device__ __forceinline__ float sigmoidf_(float v) {
    return 1.0f / (1.0f + __expf(-v));
}

// ---------------------------------------------------------------------------
// Kernel 1: partial sum/max pooling over HW chunks.
// grid (NCHK, B), block 256 (thread == channel). Lane-consecutive channels ->
// every global_load is a fully coalesced 1KB burst.
// ---------------------------------------------------------------------------
__global__ void k_pool_partial(const float* __restrict__ x,
                               float* __restrict__ psum,
                               float* __restrict__ pmax) {
    const int c     = threadIdx.x;
    const int chunk = blockIdx.x;
    const int b     = blockIdx.y;
    const float* xp = x + ((size_t)b * HWn + (size_t)chunk * CHUNK) * Cn + c;

    float s = 0.0f;
    float m = -3.402823466e38f;
    #pragma unroll 4
    for (int i = 0; i < CHUNK; ++i) {
        float v = xp[(size_t)i * Cn];
        s += v;
        m = fmaxf(m, v);
    }
    psum[((size_t)chunk * Bn + b) * Cn + c] = s;
    pmax[((size_t)chunk * Bn + b) * Cn + c] = m;
}

// ---------------------------------------------------------------------------
// Kernel 2: finish pooling, then the whole channel MLP with f32 WMMA.
// Single block of 256 threads (8 waves).
//   GEMM1: h = relu(P[32,256] @ w1[256,16] + b1)   (P = avg or max pool)
//   GEMM2: s = sigmoid(h[32,16] @ w2[16,256] + b2) ; chout = s_avg + s_max
// A-fragment layout (16x4 f32): lanes 0-15 hold K={k0,k0+1}, lanes 16-31 hold
// K={k0+2,k0+3}, row M = lane&15. B/C/D: row striped across lanes.
// ---------------------------------------------------------------------------
__global__ void k_channel_mlp(const float* __restrict__ psum,
                              const float* __restrict__ pmax,
                              const float* __restrict__ w1,
                              const float* __restrict__ b1,
                              const float* __restrict__ w2,
                              const float* __restrict__ b2,
                              float* __restrict__ chout) {
    __shared__ float lds_pool[2][Bn][Cn + 1];   // +1 pad: bank-conflict free
    __shared__ float lds_h[2][Bn][HID + 1];

    const int tid  = threadIdx.x;
    const int wv   = tid >> 5;
    const int lane = tid & 31;

    // ---- reduce the NCHK partials; avg = sum / HW ----
    for (int p = tid; p < Bn * Cn; p += 256) {
        const int b = p >> 8, c = p & (Cn - 1);
        float s = 0.0f, m = -3.402823466e38f;
        #pragma unroll
        for (int ch = 0; ch < NCHK; ++ch) {
            s += psum[((size_t)ch * Bn + b) * Cn + c];
            m = fmaxf(m, pmax[((size_t)ch * Bn + b) * Cn + c]);
        }
        lds_pool[0][b][c] = s * (1.0f / (float)HWn);
        lds_pool[1][b][c] = m;
    }
    __syncthreads();

    // ---- GEMM1: waves 0..3 each own one (source, m-tile) 16x16 output ----
    if (wv < 4) {
        const int src  = wv >> 1;
        const int mt   = wv & 1;
        const int col  = lane & 15;
        const int m    = mt * 16 + col;          // A row for this lane
        const int ksel = (lane < 16) ? 0 : 2;

        v8f acc = {};
        for (int k0 = 0; k0 < Cn; k0 += 4) {
            v2f a, bfr;
            a.x   = lds_pool[src][m][k0 + ksel];
            a.y   = lds_pool[src][m][k0 + ksel + 1];
            bfr.x = w1[(k0 + ksel)     * HID + col];
            bfr.y = w1[(k0 + ksel + 1) * HID + col];
            acc = wmma4(a, bfr, acc);
        }
        const float bias  = b1[col];
        const int   rbase = mt * 16 + ((lane & 16) ? 8 : 0);
        #pragma unroll
        for (int v = 0; v < 8; ++v)
            lds_h[src][rbase + v][col] = fmaxf(acc[v] + bias, 0.0f);
    }
    __syncthreads();

    // ---- GEMM2: 32 (m-tile, n-tile) outputs over 8 waves; sum both sigmoids
    for (int t = wv; t < 32; t += 8) {
        const int mt   = t & 1;
        const int nt   = t >> 1;
        const int col  = lane & 15;
        const int ksel = (lane < 16) ? 0 : 2;
        const float bias = b2[nt * 16 + col];

        float combined[8];
        #pragma unroll
        for (int v = 0; v < 8; ++v) combined[v] = 0.0f;

        #pragma unroll
        for (int src = 0; src < 2; ++src) {
            v8f acc = {};
            #pragma unroll
            for (int k0 = 0; k0 < HID; k0 += 4) {
                v2f a, bfr;
                a.x   = lds_h[src][mt * 16 + col][k0 + ksel];
                a.y   = lds_h[src][mt * 16 + col][k0 + ksel + 1];
                bfr.x = w2[(k0 + ksel)     * Cn + nt * 16 + col];
                bfr.y = w2[(k0 + ksel + 1) * Cn + nt * 16 + col];
                acc = wmma4(a, bfr, acc);
            }
            #pragma unroll
            for (int v = 0; v < 8; ++v)
                combined[v] += sigmoidf_(acc[v] + bias);
        }
        const int rbase = mt * 16 + ((lane & 16) ? 8 : 0);
        #pragma unroll
        for (int v = 0; v < 8; ++v)
            chout[(size_t)(rbase + v) * Cn + nt * 16 + col] = combined[v];
    }
}

// ---------------------------------------------------------------------------
// Kernel 3: spatial stats of xg = x * chout : mean & max over C per pixel.
// One wave per pixel; lane l covers channels l, l+32, ..., l+224, then a
// 5-step xor-shuffle reduction (wave32).
// ---------------------------------------------------------------------------
__global__ void k_spatial_stats(const float* __restrict__ x,
                                const float* __restrict__ chout,
                                float* __restrict__ sp) {
    const int lane = threadIdx.x & 31;
    const int wv   = threadIdx.x >> 5;
    const int pos  = blockIdx.x * 8 + wv;      // [0, B*HW)
    const int b    = pos / HWn;

    const float* xp = x + (size_t)pos * Cn;
    const float* cp = chout + (size_t)b * Cn;

    float s = 0.0f, m = -3.402823466e38f;
    #pragma unroll
    for (int i = 0; i < Cn / 32; ++i) {
        const int c = lane + i * 32;
        const float v = xp[c] * cp[c];
        s += v;
        m = fmaxf(m, v);
    }
    #pragma unroll
    for (int off = 16; off >= 1; off >>= 1) {
        s += __shfl_xor(s, off, 32);
        m = fmaxf(m, __shfl_xor(m, off, 32));
    }
    if (lane == 0) {
        sp[(size_t)pos * 2 + 0] = s * (1.0f / (float)Cn);
        sp[(size_t)pos * 2 + 1] = m;
    }
}

// ---------------------------------------------------------------------------
// Kernel 4: 7x7 SAME conv over sp[B,H,W,2] -> sigmoid -> satt[B,H,W].
// One thread per pixel; ~10M MACs total, trivial.
// ---------------------------------------------------------------------------
__global__ void k_spatial_conv(const float* __restrict__ sp,
                               const float* __restrict__ wconv,
                               const float* __restrict__ bconv,
                               float* __restrict__ satt) {
    const int idx = blockIdx.x * blockDim.x + threadIdx.x;  // [0, B*HW)
    const int b  = idx / HWn;
    const int hw = idx - b * HWn;
    const int y  = hw / Wn;
    const int xx = hw - y * Wn;

    float acc = 0.0f;
    #pragma unroll
    for (int ky = 0; ky < 7; ++ky) {
        const int iy = y + ky - 3;
        if (iy < 0 || iy >= Hn) continue;
        #pragma unroll
        for (int kx = 0; kx < 7; ++kx) {
            const int ix = xx + kx - 3;
            if (ix < 0 || ix >= Wn) continue;
            const size_t sidx = ((size_t)(b * Hn + iy) * Wn + ix) * 2;
            acc += sp[sidx + 0] * wconv[(ky * 7 + kx) * 2 + 0];
            acc += sp[sidx + 1] * wconv[(ky * 7 + kx) * 2 + 1];
        }
    }
    satt[idx] = sigmoidf_(acc + bconv[0]);
}

// ---------------------------------------------------------------------------
// Kernel 5: out = x * chout[b,c] * satt[b,h,w], float4 vectorized along C.
// ---------------------------------------------------------------------------
__global__ void k_apply(const float* __restrict__ x,
                        const float* __restrict__ chout,
                        const float* __restrict__ satt,
                        float* __restrict__ out) {
    const size_t gid = (size_t)blockIdx.x * blockDim.x + threadIdx.x;
    const size_t pos = gid >> 6;               // 64 float4 groups per pixel
    const int    c4  = (int)(gid & 63);
    const int    b   = (int)(pos / HWn);

    const float4 xv = ((const float4*)x)[gid];
    const float4 cv = ((const float4*)chout)[(size_t)b * (Cn / 4) + c4];
    const float  s  = satt[pos];

    float4 o;
    o.x = xv.x * cv.x * s;
    o.y = xv.y * cv.y * s;
    o.z = xv.z * cv.z * s;
    o.w = xv.w * cv.w * s;
    ((float4*)out)[gid] = o;
}

// ---------------------------------------------------------------------------
extern "C" void kernel_launch(void* const* d_in, const int* in_sizes, int n_in,
                              void* d_out, int out_size, void* d_ws, size_t ws_size,
                              hipStream_t stream) {
    const float* x     = (const float*)d_in[0];
    const float* w1    = (const float*)d_in[1];
    const float* b1    = (const float*)d_in[2];
    const float* w2    = (const float*)d_in[3];
    const float* b2    = (const float*)d_in[4];
    const float* wconv = (const float*)d_in[5];
    const float* bconv = (const float*)d_in[6];
    float* out = (float*)d_out;

    // workspace layout (floats): ~1.7 MB total, lives in L2
    float* ws    = (float*)d_ws;
    float* psum  = ws;                                   // [NCHK,B,C]
    float* pmax  = psum + (size_t)NCHK * Bn * Cn;        // [NCHK,B,C]
    float* chout = pmax + (size_t)NCHK * Bn * Cn;        // [B,C]
    float* sp    = chout + (size_t)Bn * Cn;              // [B,H,W,2]
    float* satt  = sp + (size_t)Bn * HWn * 2;            // [B,H,W]

    // 1) partial pooling over HW
    k_pool_partial<<<dim3(NCHK, Bn), 256, 0, stream>>>(x, psum, pmax);

    // 2) finish pooling + WMMA channel MLP (single workgroup)
    k_channel_mlp<<<1, 256, 0, stream>>>(psum, pmax, w1, b1, w2, b2, chout);

    // 3) spatial mean/max of xg (wave per pixel)
    k_spatial_stats<<<(Bn * HWn) / 8, 256, 0, stream>>>(x, chout, sp);

    // 4) 7x7 conv + sigmoid
    k_spatial_conv<<<(Bn * HWn) / 256, 256, 0, stream>>>(sp, wconv, bconv, satt);

    // 5) fused final scaling
    k_apply<<<(Bn * HWn * Cn / 4) / 256, 256, 0, stream>>>(x, chout, satt, out);
}